// MorphModule_36919538876697
// MI455X (gfx1250) — compile-verified
//
#include <hip/hip_runtime.h>

#define IMG_H 4096
#define IMG_W 6144
#define TILE_H 32
#define TILE_W 128
#define LDS_W 136        // TILE_W + 4-float halo each side (16B-aligned chunks)
#define LDS_ROWS 34      // TILE_H + 1 halo row each side
#define NCHUNK 34        // LDS_W / 4  (16-byte chunks per LDS row)
#define BORDER_VAL 1e20f
#define NTHREADS 256

// ---- CDNA5 async global->LDS copy (ASYNCcnt-tracked), 16 bytes per lane ----
// GV addressing mode: vdst = LDS byte offset VGPR, vaddr = 64-bit global
// address VGPR pair, saddr = off (NULL). Generic pointers to LDS carry the
// LDS byte offset in their low 32 bits (aperture check uses only addr[63:32]).
__device__ __forceinline__ void async_copy16(const float* g, float* l) {
  asm volatile("global_load_async_to_lds_b128 %0, %1, off"
               :
               : "v"((unsigned)(unsigned long long)l),
                 "v"((unsigned long long)g)
               : "memory");
}

__device__ __forceinline__ void wait_async_zero() {
  asm volatile("s_wait_asynccnt 0" ::: "memory");
}

__global__ __launch_bounds__(NTHREADS)
void erode3x3_async_kernel(const float* __restrict__ img,
                           const float* __restrict__ se,
                           float* __restrict__ out) {
  __shared__ float tile[LDS_ROWS * LDS_W];

  const int tid   = threadIdx.x;
  const int tileX = blockIdx.x * TILE_W;   // multiple of 128 (and of 4)
  const int tileY = blockIdx.y * TILE_H;

  // ---------------- Stage (TILE + halo) into LDS ----------------
  // LDS col c maps to global col tileX - 4 + c; LDS row r -> global row tileY - 1 + r.
  const bool interior = (tileX > 0) && (tileX + TILE_W < IMG_W) &&
                        (tileY > 0) && (tileY + TILE_H < IMG_H);
  if (interior) {
    // Hot path (~94% of blocks): every chunk in-bounds -> pure async DMA.
    const float* gbase = img + (size_t)(tileY - 1) * IMG_W + (tileX - 4);
#pragma unroll 1
    for (int idx = tid; idx < LDS_ROWS * NCHUNK; idx += NTHREADS) {
      const int row = idx / NCHUNK;
      const int ch  = idx - row * NCHUNK;
      async_copy16(gbase + (size_t)row * IMG_W + ch * 4,
                   &tile[row * LDS_W + ch * 4]);
    }
  } else {
#pragma unroll 1
    for (int idx = tid; idx < LDS_ROWS * NCHUNK; idx += NTHREADS) {
      const int row = idx / NCHUNK;
      const int ch  = idx - row * NCHUNK;
      const int gr  = tileY - 1 + row;
      const int gc  = tileX - 4 + ch * 4;
      float* lp = &tile[row * LDS_W + ch * 4];
      if (gr >= 0 && gr < IMG_H && gc >= 0 && (gc + 3) < IMG_W) {
        async_copy16(img + (size_t)gr * IMG_W + gc, lp);
      } else {
        // border chunk: per-element guarded fill (OOB -> +1e20 so it never wins)
        float4 v;
        const bool rowok = (gr >= 0 && gr < IMG_H);
        v.x = (rowok && (gc + 0) >= 0 && (gc + 0) < IMG_W) ? img[(size_t)gr * IMG_W + gc + 0] : BORDER_VAL;
        v.y = (rowok && (gc + 1) >= 0 && (gc + 1) < IMG_W) ? img[(size_t)gr * IMG_W + gc + 1] : BORDER_VAL;
        v.z = (rowok && (gc + 2) >= 0 && (gc + 2) < IMG_W) ? img[(size_t)gr * IMG_W + gc + 2] : BORDER_VAL;
        v.w = (rowok && (gc + 3) >= 0 && (gc + 3) < IMG_W) ? img[(size_t)gr * IMG_W + gc + 3] : BORDER_VAL;
        *(float4*)lp = v;
      }
    }
  }

  wait_async_zero();   // drain ASYNCcnt (async LDS writes visible)
  __syncthreads();     // covers the DS-store fallback path + workgroup sync

  // ---------------- Structuring element in registers ----------------
  float s[3][3];
#pragma unroll
  for (int p = 0; p < 3; ++p)
#pragma unroll
    for (int q = 0; q < 3; ++q)
      s[p][q] = se[p * 3 + q];

  // ---------------- Compute: 4 x float4 outputs per thread ----------------
  const int tx = tid & 31;   // 0..31 -> 4-wide column group
  const int ty = tid >> 5;   // 0..7
  const int lc = tx * 4;     // local output col (0..124)

#pragma unroll
  for (int lr = ty; lr < TILE_H; lr += 8) {
    float o0 = BORDER_VAL, o1 = BORDER_VAL, o2 = BORDER_VAL, o3 = BORDER_VAL;
#pragma unroll
    for (int p = 0; p < 3; ++p) {
      // output col lc maps to LDS col lc+4; taps need cols (lc+4)-1 .. (lc+4+3)+1
      const float* rp = &tile[(lr + p) * LDS_W + lc + 3];
      const float v0 = rp[0], v1 = rp[1], v2 = rp[2];
      const float v3 = rp[3], v4 = rp[4], v5 = rp[5];
      o0 = fminf(o0, fminf(fminf(v0 - s[p][0], v1 - s[p][1]), v2 - s[p][2]));
      o1 = fminf(o1, fminf(fminf(v1 - s[p][0], v2 - s[p][1]), v3 - s[p][2]));
      o2 = fminf(o2, fminf(fminf(v2 - s[p][0], v3 - s[p][1]), v4 - s[p][2]));
      o3 = fminf(o3, fminf(fminf(v3 - s[p][0], v4 - s[p][1]), v5 - s[p][2]));
    }
    float4 r = make_float4(o0, o1, o2, o3);
    *(float4*)&out[(size_t)(tileY + lr) * IMG_W + tileX + lc] = r;  // 16B aligned
  }
}

extern "C" void kernel_launch(void* const* d_in, const int* in_sizes, int n_in,
                              void* d_out, int out_size, void* d_ws, size_t ws_size,
                              hipStream_t stream) {
  const float* img = (const float*)d_in[0];   // [4096, 6144] f32
  const float* se  = (const float*)d_in[1];   // [3, 3] f32
  float* out = (float*)d_out;                 // [4096, 6144] f32

  dim3 grid(IMG_W / TILE_W, IMG_H / TILE_H);  // 48 x 128 tiles, exact cover
  dim3 block(NTHREADS);
  erode3x3_async_kernel<<<grid, block, 0, stream>>>(img, se, out);
}